// SuSPP_20813411516761
// MI455X (gfx1250) — compile-verified
//
#include <hip/hip_runtime.h>
#include <math.h>

typedef __attribute__((ext_vector_type(2))) float v2f;
typedef __attribute__((ext_vector_type(4))) float v4f;
typedef __attribute__((ext_vector_type(8))) float v8f;

#define NPTS 4096
#define DIMS 64
#define A_LOG (-8.317766166719343f)   // -log(4096)
#define BLUR2 (1e-6f)
#define MAX_ITERS 160

// ---------------------------------------------------------------------------
// 1) diameter + eps schedule (device-side, since schedule length is data-dep)
// ---------------------------------------------------------------------------
__global__ void sched_kernel(const float* __restrict__ x, const float* __restrict__ y,
                             float* __restrict__ eps_arr, int* __restrict__ schedLen) {
  __shared__ float smin[256], smax[256];
  const int tid = threadIdx.x;
  const int d = tid & 63;
  const int chunk = tid >> 6;                   // 4 chunks of 2048 combined rows
  const float* base = (chunk < 2) ? x : y;
  const int r0 = (chunk & 1) * 2048;
  float mn = __builtin_inff(), mx = -__builtin_inff();
  for (int r = r0; r < r0 + 2048; ++r) {
    float v = base[(size_t)r * DIMS + d];
    mn = fminf(mn, v); mx = fmaxf(mx, v);
  }
  smin[tid] = mn; smax[tid] = mx;
  __syncthreads();
  if (tid < 64) {
    mn = fminf(fminf(smin[tid], smin[tid + 64]), fminf(smin[tid + 128], smin[tid + 192]));
    mx = fmaxf(fmaxf(smax[tid], smax[tid + 64]), fmaxf(smax[tid + 128], smax[tid + 192]));
    float r = mx - mn;
    smin[tid] = r * r;
  }
  __syncthreads();
  if (tid == 0) {
    float d2 = 0.f;                              // diameter^2 = eps0 (p=2)
    for (int k = 0; k < 64; ++k) d2 += smin[k];
    int idx = 0;
    eps_arr[idx++] = d2;
    const float start = logf(d2);                // 2*ln(diameter)
    const float stop  = -13.815511f;             // 2*ln(0.001)
    const float step  = -0.21072103f;            // 2*ln(0.9)
    for (int k = 0; idx < 250; ++k) {            // numpy arange semantics
      float v = start + (float)k * step;
      if (!(v > stop)) break;
      eps_arr[idx++] = expf(v);
    }
    eps_arr[idx++] = BLUR2;
    *schedLen = idx;
  }
}

// ---------------------------------------------------------------------------
// 2) squared norms of x rows and y rows
// ---------------------------------------------------------------------------
__global__ void norms_kernel(const float* __restrict__ x, const float* __restrict__ y,
                             float* __restrict__ xsq, float* __restrict__ ysq) {
  __shared__ float s[64];
  int row = blockIdx.x;
  const float* p;
  float* outp;
  int r;
  if (row < NPTS) { p = x + (size_t)row * DIMS; outp = xsq; r = row; }
  else            { p = y + (size_t)(row - NPTS) * DIMS; outp = ysq; r = row - NPTS; }
  float v = p[threadIdx.x];
  s[threadIdx.x] = v * v;
  __syncthreads();
  for (int off = 32; off > 0; off >>= 1) {
    if (threadIdx.x < off) s[threadIdx.x] += s[threadIdx.x + off];
    __syncthreads();
  }
  if (threadIdx.x == 0) outp[r] = s[0];
}

// ---------------------------------------------------------------------------
// 3) cost matrix via fp32 WMMA: C_ij = 0.5(|x_i|^2+|y_j|^2) - x_i.y_j ; also C^T
//    One wave per 16x16 tile; 16 x V_WMMA_F32_16X16X4_F32 over K=64.
// ---------------------------------------------------------------------------
__global__ void cost_wmma_kernel(const float* __restrict__ x, const float* __restrict__ y,
                                 const float* __restrict__ xsq, const float* __restrict__ ysq,
                                 float* __restrict__ C, float* __restrict__ CT) {
  const int lane   = threadIdx.x;              // wave32
  const int tile_j = blockIdx.x * 4 + threadIdx.y;
  const int tile_i = blockIdx.y;
  const int half   = lane >> 4;                // 0: lanes 0-15, 1: lanes 16-31
  const int l15    = lane & 15;
  // A(16x4 f32): lane holds M=l15, VGPR0->K=2*half, VGPR1->K=2*half+1
  const float* xrow = x + (size_t)(tile_i * 16 + l15) * DIMS + 2 * half;
  // B(4x16 f32): lane holds N=l15, VGPR0->K=2*half, VGPR1->K=2*half+1 (B = y^T tile)
  const float* yrow = y + (size_t)(tile_j * 16 + l15) * DIMS + 2 * half;

  v8f acc = {0.f, 0.f, 0.f, 0.f, 0.f, 0.f, 0.f, 0.f};
#pragma unroll
  for (int kk = 0; kk < 16; ++kk) {
    v2f a = *(const v2f*)(xrow + kk * 4);
    v2f b = *(const v2f*)(yrow + kk * 4);
    acc = __builtin_amdgcn_wmma_f32_16x16x4_f32(false, a, false, b,
                                                (short)0, acc, false, false);
  }
#pragma unroll
  for (int r = 0; r < 8; ++r) {
    // D layout: VGPR r -> M = r + 8*half, N = l15
    int i = tile_i * 16 + r + 8 * half;
    int j = tile_j * 16 + l15;
    float val = 0.5f * (xsq[i] + ysq[j]) - acc[r];
    C [(size_t)i * NPTS + j] = val;
    CT[(size_t)j * NPTS + i] = val;
  }
}

// ---------------------------------------------------------------------------
// 4) fused softmin. Convention for t:
//    t == -1 : init step   (eps = eps_arr[0], potential treated as 0)
//    t >= 0  : loop iter t (early-out if t >= schedLen)
//    t == -2 : final step  (eps = blur^2)
// out[row] = -eps * logsumexp_j( a_log + (pot[j] - Cmat[row,j]) / eps )
// Two-phase (register-staged): block max, then exactly ONE exp per element.
// ---------------------------------------------------------------------------
__global__ void softmin_kernel(const float* __restrict__ Cmat, const float* __restrict__ pot,
                               float* __restrict__ out, const float* __restrict__ eps_arr,
                               const int* __restrict__ schedLen, int t) {
  if (t >= 0 && t >= *schedLen) return;
  const float eps = (t == -2) ? BLUR2 : eps_arr[(t < 0) ? 0 : t];
  const float inv_eps = 1.0f / eps;
  const int row = blockIdx.x;
  const int tid = threadIdx.x;
  const float* Crow = Cmat + (size_t)row * NPTS;

  float v[16];
  float m = -__builtin_inff();
#pragma unroll
  for (int s = 0; s < 4; ++s) {
    const int j = s * 1024 + tid * 4;            // global_load_b128 per lane
    const v4f c = *(const v4f*)(Crow + j);
    v4f p;
    if (t == -1) p = (v4f){0.f, 0.f, 0.f, 0.f};  // uniform branch
    else         p = *(const v4f*)(pot + j);
#pragma unroll
    for (int q = 0; q < 4; ++q) {
      float val = A_LOG + (p[q] - c[q]) * inv_eps;
      v[s * 4 + q] = val;
      m = fmaxf(m, val);
    }
  }

  __shared__ float red[256];
  red[tid] = m;
  __syncthreads();
  for (int off = 128; off > 0; off >>= 1) {
    if (tid < off) red[tid] = fmaxf(red[tid], red[tid + off]);
    __syncthreads();
  }
  const float M = red[0];
  __syncthreads();

  float ssum = 0.f;
#pragma unroll
  for (int k = 0; k < 16; ++k) ssum += __expf(v[k] - M);   // 1 exp / element
  red[tid] = ssum;
  __syncthreads();
  for (int off = 128; off > 0; off >>= 1) {
    if (tid < off) red[tid] += red[tid + off];
    __syncthreads();
  }
  if (tid == 0) out[row] = -eps * (M + __logf(red[0]));
}

__global__ void combine_kernel(float* __restrict__ f, float* __restrict__ g,
                               const float* __restrict__ ft, const float* __restrict__ gt,
                               const float* __restrict__ eps_arr,
                               const int* __restrict__ schedLen, int t) {
  if (t >= 0 && t >= *schedLen) return;
  int j = blockIdx.x * blockDim.x + threadIdx.x;
  if (j >= NPTS) return;
  float eps = eps_arr[(t < 0) ? 0 : t];
  float damp = 1.0f / (1.0f + eps);            // rho = reach^p = 1
  if (t == -1) { f[j] = ft[j]; g[j] = damp * gt[j]; }
  else         { f[j] = 0.5f * (f[j] + ft[j]); g[j] = 0.5f * (g[j] + damp * gt[j]); }
}

// out = <a,f> + (rho + eps/2) * <b, 1 - exp(-g/rho)>, with g = damp(eps)*gt_raw
__global__ void reduce_kernel(const float* __restrict__ ff, const float* __restrict__ gg,
                              float* __restrict__ out) {
  __shared__ float s1[256], s2[256];
  const float eps  = BLUR2;
  const float damp = 1.0f / (1.0f + eps);
  float af = 0.f, bg = 0.f;
  for (int j = threadIdx.x; j < NPTS; j += 256) {
    af += ff[j];
    float gj = damp * gg[j];                   // apply damp to final g
    bg += 1.0f - __expf(-gj);                  // rho = 1
  }
  s1[threadIdx.x] = af; s2[threadIdx.x] = bg;
  __syncthreads();
  for (int off = 128; off > 0; off >>= 1) {
    if (threadIdx.x < off) {
      s1[threadIdx.x] += s1[threadIdx.x + off];
      s2[threadIdx.x] += s2[threadIdx.x + off];
    }
    __syncthreads();
  }
  if (threadIdx.x == 0)
    out[0] = s1[0] * (1.0f / NPTS) + (1.0f + 0.5f * eps) * (s2[0] * (1.0f / NPTS));
}

// ---------------------------------------------------------------------------
extern "C" void kernel_launch(void* const* d_in, const int* in_sizes, int n_in,
                              void* d_out, int out_size, void* d_ws, size_t ws_size,
                              hipStream_t stream) {
  (void)in_sizes; (void)n_in; (void)out_size; (void)ws_size;
  const float* x = (const float*)d_in[0];      // fake_data     [4096,64] f32
  const float* y = (const float*)d_in[1];      // reference_img [4096,64] f32
  float* out = (float*)d_out;

  float* ws  = (float*)d_ws;
  float* Cm  = ws;                                   // 4096*4096 floats (64MB)
  float* CT  = ws + (size_t)NPTS * NPTS;             // 64MB
  float* aux = ws + 2 * (size_t)NPTS * NPTS;
  float* xsq = aux;
  float* ysq = aux + NPTS;
  float* f   = aux + 2 * NPTS;
  float* g   = aux + 3 * NPTS;
  float* ft  = aux + 4 * NPTS;
  float* gt  = aux + 5 * NPTS;
  float* eps_arr = aux + 6 * NPTS;                   // 256 entries
  int*   schedLen = (int*)(eps_arr + 256);

  sched_kernel<<<1, 256, 0, stream>>>(x, y, eps_arr, schedLen);
  norms_kernel<<<2 * NPTS, 64, 0, stream>>>(x, y, xsq, ysq);
  cost_wmma_kernel<<<dim3(NPTS / 64, NPTS / 16), dim3(32, 4), 0, stream>>>(
      x, y, xsq, ysq, Cm, CT);

  // init potentials: f = softmin(C, 0), g = damp*softmin(C^T, 0)
  softmin_kernel<<<NPTS, 256, 0, stream>>>(Cm, g, ft, eps_arr, schedLen, -1);
  softmin_kernel<<<NPTS, 256, 0, stream>>>(CT, f, gt, eps_arr, schedLen, -1);
  combine_kernel<<<NPTS / 256, 256, 0, stream>>>(f, g, ft, gt, eps_arr, schedLen, -1);

  // eps-scaling loop (fixed upper bound; kernels early-out past schedLen)
  for (int t = 0; t < MAX_ITERS; ++t) {
    softmin_kernel<<<NPTS, 256, 0, stream>>>(Cm, g, ft, eps_arr, schedLen, t);
    softmin_kernel<<<NPTS, 256, 0, stream>>>(CT, f, gt, eps_arr, schedLen, t);
    combine_kernel<<<NPTS / 256, 256, 0, stream>>>(f, g, ft, gt, eps_arr, schedLen, t);
  }

  // final extrapolation at eps = blur^2 and scalar reduction
  softmin_kernel<<<NPTS, 256, 0, stream>>>(Cm, g, ft, eps_arr, schedLen, -2);
  softmin_kernel<<<NPTS, 256, 0, stream>>>(CT, f, gt, eps_arr, schedLen, -2);
  reduce_kernel<<<1, 256, 0, stream>>>(ft, gt, out);
}